// BasicBlock_53171695125036
// MI455X (gfx1250) — compile-verified
//
#include <hip/hip_runtime.h>
#include <math.h>

#define NPTS 6144
#define CCH  96
#define BB   2
#define DD   32
#define HH   32
#define WW   32
#define KNN  16

typedef __attribute__((ext_vector_type(2))) float v2f;
typedef __attribute__((ext_vector_type(8))) float v8f;

// ============================================================================
// WMMA f32 strip GEMM: one wave computes 16 rows x (NT*16) cols of
// Y = act(X @ W^T + bias).  W is [mc x kd] row-major, mc == NT*16,
// KD % 4 == 0.  Per K-step: one A (v2f, b64) load feeds NT WMMAs;
// B loads are b64 per column tile.  Max A reuse, fully unrolled.
// A layout (16x4 f32): lane L holds row m=L%16, K = (L<16 ? {0,1} : {2,3}).
// B layout (4x16 f32): lane L holds col n=L%16, K = (L<16 ? {0,1} : {2,3}).
// D layout (16x16 f32): VGPR r -> row (r + 8*(L>=16)), col = L%16.
// ============================================================================
template<int KD, int NT>
__global__ void k_gemm_strip(const float* __restrict__ X, int ldx,
                             const float* __restrict__ Wt, int ldw,
                             const float* __restrict__ bias,
                             float* __restrict__ Y, int ldy,
                             int nr, int act)
{
  int wave = (blockIdx.x * blockDim.x + threadIdx.x) >> 5;
  int lane = threadIdx.x & 31;
  int i0 = wave << 4;
  if (i0 >= nr) return;
  int m  = lane & 15;
  int kh = (lane >> 4) << 1;           // 0 or 2
  const float* xr = X  + (size_t)(i0 + m) * ldx + kh;
  const float* wr = Wt + (size_t)m * ldw + kh;   // m doubles as n index
  v8f acc[NT];
#pragma unroll
  for (int t = 0; t < NT; ++t) acc[t] = (v8f){0.f,0.f,0.f,0.f,0.f,0.f,0.f,0.f};
#pragma unroll
  for (int k0 = 0; k0 < KD; k0 += 4) {
    v2f A = *(const v2f*)(xr + k0);
#pragma unroll
    for (int t = 0; t < NT; ++t) {
      v2f Bv = *(const v2f*)(wr + (size_t)t * 16 * ldw + k0);
      acc[t] = __builtin_amdgcn_wmma_f32_16x16x4_f32(false, A, false, Bv,
                                                     (short)0, acc[t], false, false);
    }
  }
  int rb = i0 + ((lane >> 4) << 3);
#pragma unroll
  for (int t = 0; t < NT; ++t) {
    int col = t * 16 + m;
    float bv = bias ? bias[col] : 0.f;
#pragma unroll
    for (int r = 0; r < 8; ++r) {
      float v = acc[t][r] + bv;
      if (act) v = fmaxf(v, 0.f);
      Y[(size_t)(rb + r) * ldy + col] = v;
    }
  }
}

// ============================================================================
// Implicit-GEMM sparse 3x3x3 conv, strip form: one wave computes 16 points x
// all 96 out-channels (6 accumulators) for taps [tap0, tap1).  The gathered
// grid (A) operand is loaded ONCE per K-step and feeds 6 WMMAs.  OOB taps use
// a safe fallback pointer + multiplicative mask (loads stay unconditional).
// Taps are split across 3 launches into partial buffers for occupancy;
// partials are summed deterministically afterwards.
// w layout: [dz][dy][dx][ci][co]  (DHWIO), grid layout: [b][z][y][x][c].
// ============================================================================
__global__ void k_conv_strip(const float* __restrict__ grid, const int* __restrict__ idx,
                             const float* __restrict__ w, float* __restrict__ y,
                             int tap0, int tap1)
{
  int wave = (blockIdx.x * blockDim.x + threadIdx.x) >> 5;
  int lane = threadIdx.x & 31;
  int i0 = wave << 4;
  if (i0 >= NPTS) return;
  int m  = lane & 15;
  int kh = (lane >> 4) << 1;
  int pi = i0 + m;
  int pb = idx[pi*4+0], pz = idx[pi*4+1], py = idx[pi*4+2], px = idx[pi*4+3];
  const float* wcol = w + m;             // + ((tap*CCH + k) * CCH) + t*16
  v8f acc[6];
#pragma unroll
  for (int t = 0; t < 6; ++t) acc[t] = (v8f){0.f,0.f,0.f,0.f,0.f,0.f,0.f,0.f};
  for (int tap = tap0; tap < tap1; ++tap) {
    int dz = tap / 9 - 1, dy = (tap / 3) % 3 - 1, dx = tap % 3 - 1;
    int zz = pz + dz, yy = py + dy, xx = px + dx;
    bool ok = (zz >= 0 && zz < DD && yy >= 0 && yy < HH && xx >= 0 && xx < WW);
    float msk = ok ? 1.f : 0.f;
    size_t go = ok ? ((((size_t)pb * DD + zz) * HH + yy) * WW + xx) * CCH : 0;
    const float* ap = grid + go + kh;
    const float* wp = wcol + (size_t)tap * CCH * CCH + (size_t)kh * CCH;
#pragma unroll
    for (int k0 = 0; k0 < CCH; k0 += 4) {
      v2f A = *(const v2f*)(ap + k0);
      A.x *= msk; A.y *= msk;
#pragma unroll
      for (int t = 0; t < 6; ++t) {
        v2f Bv;
        Bv.x = wp[(size_t)k0 * CCH + t*16];
        Bv.y = wp[(size_t)(k0 + 1) * CCH + t*16];
        acc[t] = __builtin_amdgcn_wmma_f32_16x16x4_f32(false, A, false, Bv,
                                                       (short)0, acc[t], false, false);
      }
    }
  }
  int rb = i0 + ((lane >> 4) << 3);
#pragma unroll
  for (int t = 0; t < 6; ++t) {
    int col = t * 16 + m;
#pragma unroll
    for (int r = 0; r < 8; ++r)
      y[(size_t)(rb + r) * CCH + col] = acc[t][r];
  }
}

// deterministic 3-way partial sum
__global__ void k_sum3(const float* __restrict__ a, const float* __restrict__ b,
                       const float* __restrict__ c, float* __restrict__ out, int n)
{
  int t = blockIdx.x * blockDim.x + threadIdx.x;
  if (t >= n) return;
  out[t] = a[t] + b[t] + c[t];
}

// ===== scalar GEMM for tiny widths (M=3/9 or K=9): thread per output element =====
__global__ void k_small_gemm(const float* __restrict__ X, int ldx,
                             const float* __restrict__ Wt, int ldw,
                             const float* __restrict__ bias,
                             float* __restrict__ Y, int ldy,
                             int nr, int mc, int kd)
{
  int t = blockIdx.x * blockDim.x + threadIdx.x;
  if (t >= nr * mc) return;
  int i = t / mc, j = t - i * mc;
  float s = bias ? bias[j] : 0.f;
  for (int k = 0; k < kd; ++k) s += X[(size_t)i*ldx + k] * Wt[(size_t)j*ldw + k];
  Y[(size_t)i*ldy + j] = s;
}

// ===== column mean/var (population) over nr rows, one block per column =====
__global__ void k_colstats(const float* __restrict__ Z, int ld, int nr, int mc,
                           float* __restrict__ mean, float* __restrict__ var)
{
  int c = blockIdx.x;
  __shared__ float ss[256], s2[256];
  float a = 0.f, b = 0.f;
  for (int i = threadIdx.x; i < nr; i += blockDim.x) {
    float v = Z[(size_t)i*ld + c]; a += v; b += v*v;
  }
  ss[threadIdx.x] = a; s2[threadIdx.x] = b; __syncthreads();
  for (int st = 128; st > 0; st >>= 1) {
    if (threadIdx.x < st) { ss[threadIdx.x]+=ss[threadIdx.x+st]; s2[threadIdx.x]+=s2[threadIdx.x+st]; }
    __syncthreads();
  }
  if (threadIdx.x == 0) {
    float mu = ss[0] / nr;
    mean[c] = mu;
    var[c]  = fmaxf(s2[0] / nr - mu*mu, 0.f);
  }
}

// ===== BN + optional pre-residual + optional relu + optional post-residual =====
__global__ void k_bn_act(const float* __restrict__ Z, const float* __restrict__ mean,
                         const float* __restrict__ var, const float* __restrict__ g,
                         const float* __restrict__ be, const float* __restrict__ rpre,
                         const float* __restrict__ rpost, float* __restrict__ out,
                         int total, int mc, int act)
{
  int t = blockIdx.x * blockDim.x + threadIdx.x;
  if (t >= total) return;
  int c = t % mc;
  float v = (Z[t] - mean[c]) * rsqrtf(var[c] + 1e-5f) * g[c] + be[c];
  if (rpre)  v += rpre[t];
  if (act)   v = fmaxf(v, 0.f);
  if (rpost) v += rpost[t];
  out[t] = v;
}

// ===== attention enhance: enh = [cd, cl*att0(cl), nm*att1(nm)] =====
__global__ void k_att_enh(const float* __restrict__ p, const float* __restrict__ aw1,
                          const float* __restrict__ ab1, const float* __restrict__ aw2,
                          const float* __restrict__ ab2, float* __restrict__ enh)
{
  int i = blockIdx.x * blockDim.x + threadIdx.x;
  if (i >= NPTS) return;
  const float* pi = p + (size_t)i*9;
  float o[9];
  o[0]=pi[0]; o[1]=pi[1]; o[2]=pi[2];
  for (int a = 0; a < 2; ++a) {
    const float* f  = pi + 3 + a*3;
    const float* w1 = aw1 + a*48;
    const float* b1 = ab1 + a*16;
    const float* w2 = aw2 + a*48;
    const float* b2 = ab2 + a*3;
    float h[16];
    for (int j = 0; j < 16; ++j) {
      float s = b1[j];
      for (int k = 0; k < 3; ++k) s += w1[j*3+k]*f[k];
      h[j] = fmaxf(s, 0.f);
    }
    for (int j = 0; j < 3; ++j) {
      float s = b2[j];
      for (int k = 0; k < 16; ++k) s += w2[j*16+k]*h[k];
      float sg = 1.f / (1.f + expf(-s));
      o[3 + a*3 + j] = f[j] * sg;
    }
  }
  for (int j = 0; j < 9; ++j) enh[(size_t)i*9 + j] = o[j];
}

// ===== sem gate + feature blend =====
__global__ void k_sem_feat(const float* __restrict__ h48, const float* __restrict__ sw2,
                           const float* __restrict__ sb2, const float* __restrict__ fu,
                           const float* __restrict__ x, float* __restrict__ feat)
{
  int i = blockIdx.x * blockDim.x + threadIdx.x;
  if (i >= NPTS) return;
  float s = sb2[0];
  for (int k = 0; k < 48; ++k) s += h48[(size_t)i*48 + k] * sw2[k];
  float sem = 1.f / (1.f + expf(-s));
  for (int c = 0; c < CCH; ++c) {
    size_t t = (size_t)i*CCH + c;
    feat[t] = fu[t]*sem + x[t]*(1.f - sem);
  }
}

// ===== KNN (top-16 same-batch) + 3x3 eigen geometry -> (linearity, density) =====
__global__ void k_knn(const int* __restrict__ idx, float* __restrict__ lind)
{
  int i = blockIdx.x * blockDim.x + threadIdx.x;
  if (i >= NPTS) return;
  int bi = idx[i*4];
  float cx=(float)idx[i*4+1], cy=(float)idx[i*4+2], cz=(float)idx[i*4+3];
  float best[KNN]; int bj[KNN];
  for (int k = 0; k < KNN; ++k) { best[k] = 3.0e38f; bj[k] = i; }
  for (int j = 0; j < NPTS; ++j) {
    if (j == i || idx[j*4] != bi) continue;
    float dx = cx - (float)idx[j*4+1];
    float dy = cy - (float)idx[j*4+2];
    float dz = cz - (float)idx[j*4+3];
    float d2 = dx*dx + dy*dy + dz*dz;
    if (d2 < best[KNN-1]) {
      int k = KNN-1;
      while (k > 0 && best[k-1] > d2) { best[k]=best[k-1]; bj[k]=bj[k-1]; --k; }
      best[k] = d2; bj[k] = j;
    }
  }
  float nx[KNN], ny[KNN], nz[KNN];
  float mx=0,my=0,mz=0, md=0;
  for (int k = 0; k < KNN; ++k) {
    int j = bj[k];
    nx[k]=(float)idx[j*4+1]; ny[k]=(float)idx[j*4+2]; nz[k]=(float)idx[j*4+3];
    mx+=nx[k]; my+=ny[k]; mz+=nz[k];
    md += sqrtf(fmaxf(best[k], 1e-12f));
  }
  const float ik = 1.f/KNN;
  mx*=ik; my*=ik; mz*=ik; md*=ik;
  float cxx=0,cxy=0,cxz=0,cyy=0,cyz=0,czz=0;
  for (int k = 0; k < KNN; ++k) {
    float ax=nx[k]-mx, ay=ny[k]-my, az=nz[k]-mz;
    cxx+=ax*ax; cxy+=ax*ay; cxz+=ax*az; cyy+=ay*ay; cyz+=ay*az; czz+=az*az;
  }
  const float ivk = 1.f/(KNN-1);
  cxx*=ivk; cxy*=ivk; cxz*=ivk; cyy*=ivk; cyz*=ivk; czz*=ivk;
  float e0, e1, e2;
  float p1 = cxy*cxy + cxz*cxz + cyz*cyz;
  float q  = (cxx+cyy+czz) * (1.f/3.f);
  float p2 = (cxx-q)*(cxx-q)+(cyy-q)*(cyy-q)+(czz-q)*(czz-q) + 2.f*p1;
  if (p2 < 1e-18f) { e0=e1=e2=q; }
  else {
    float pp = sqrtf(p2/6.f);
    float iv = 1.f/pp;
    float b00=(cxx-q)*iv, b11=(cyy-q)*iv, b22=(czz-q)*iv;
    float b01=cxy*iv, b02=cxz*iv, b12=cyz*iv;
    float det = b00*(b11*b22-b12*b12) - b01*(b01*b22-b12*b02) + b02*(b01*b12-b11*b02);
    float r = fminf(fmaxf(det*0.5f, -1.f), 1.f);
    float phi = acosf(r)*(1.f/3.f);
    e0 = q + 2.f*pp*cosf(phi);
    e2 = q + 2.f*pp*cosf(phi + 2.0943951023931953f);
    e1 = 3.f*q - e0 - e2;
  }
  float inv = 1.f/(e0+e1+e2+1e-6f);
  lind[i*2]   = e0*inv - (e1+e2)*inv;
  lind[i*2+1] = 1.f/(md + 1e-6f);
}

__global__ void k_coord_min(const int* __restrict__ idx, float* __restrict__ startv)
{
  __shared__ int sm[3][256];
  int m0=0x7fffffff, m1=0x7fffffff, m2=0x7fffffff;
  for (int i = threadIdx.x; i < NPTS; i += blockDim.x) {
    m0 = min(m0, idx[i*4+1]); m1 = min(m1, idx[i*4+2]); m2 = min(m2, idx[i*4+3]);
  }
  sm[0][threadIdx.x]=m0; sm[1][threadIdx.x]=m1; sm[2][threadIdx.x]=m2;
  __syncthreads();
  for (int s = 128; s > 0; s >>= 1) {
    if (threadIdx.x < s) {
      sm[0][threadIdx.x]=min(sm[0][threadIdx.x],sm[0][threadIdx.x+s]);
      sm[1][threadIdx.x]=min(sm[1][threadIdx.x],sm[1][threadIdx.x+s]);
      sm[2][threadIdx.x]=min(sm[2][threadIdx.x],sm[2][threadIdx.x+s]);
    }
    __syncthreads();
  }
  if (threadIdx.x == 0) {
    startv[0]=(float)sm[0][0]; startv[1]=(float)sm[1][0]; startv[2]=(float)sm[2][0];
  }
}

// ===== softmax(fl) -> tower/backg/line -> grid_sizes, gm =====
__global__ void k_gridsizes(const float* __restrict__ fl, const float* __restrict__ lind,
                            float* __restrict__ gs, float* __restrict__ gm)
{
  int i = blockIdx.x * blockDim.x + threadIdx.x;
  if (i >= NPTS) return;
  float l0=fl[i*3], l1=fl[i*3+1], l2=fl[i*3+2];
  float m = fmaxf(l0, fmaxf(l1, l2));
  float e0=expf(l0-m), e1=expf(l1-m), e2=expf(l2-m);
  float inv = 1.f/(e0+e1+e2);
  float p0=e0*inv, p1=e1*inv, p2=e2*inv;
  float lin = lind[i*2], den = lind[i*2+1];
  float tower = (den*2.f + p0) * (1.f/3.f);
  float backg = (fmaxf(1.f-lin, 1.f-den) + p1) * (1.f/3.f);
  float line  = (lin*2.f + p2) * (1.f/3.f);
  float g01 = tower*0.1f + backg*0.4f + line*0.2f + 1e-6f;   // dims 0,1 (lg=0.2)
  float g2  = tower*0.1f + backg*0.4f + line*1.0f + 1e-6f;   // dim 2   (lg=1.0)
  gs[i*3]=g01; gs[i*3+1]=g01; gs[i*3+2]=g2;
  gm[i] = (g01+g01+g2)*(1.f/3.f);
}

// stable rank (matches stable argsort)
__global__ void k_rank(const float* __restrict__ gm, int* __restrict__ rnk)
{
  int i = blockIdx.x * blockDim.x + threadIdx.x;
  if (i >= NPTS) return;
  float v = gm[i];
  int r = 0;
  for (int j = 0; j < NPTS; ++j) {
    float u = gm[j];
    r += (u < v) || (u == v && j < i);
  }
  rnk[i] = r;
}

__global__ void k_reps(const float* __restrict__ gs, const int* __restrict__ rnk,
                       float* __restrict__ reps)
{
  int i = blockIdx.x * blockDim.x + threadIdx.x;
  if (i >= NPTS) return;
  int r = rnk[i];
  const float sc = 0.01f;
  if (r >= 100 && r < 200) for (int d=0;d<3;++d) atomicAdd(&reps[d],   gs[i*3+d]*sc);
  if (r >= NPTS - 100)     for (int d=0;d<3;++d) atomicAdd(&reps[3+d], gs[i*3+d]*sc);
  if (r < 100)             for (int d=0;d<3;++d) atomicAdd(&reps[6+d], gs[i*3+d]*sc);
}

__global__ void k_voxel(const int* __restrict__ idx, const float* __restrict__ startv,
                        const float* __restrict__ reps, int branch,
                        int* __restrict__ cvox, int* __restrict__ mx)
{
  int i = blockIdx.x * blockDim.x + threadIdx.x;
  if (i >= NPTS) return;
  for (int d = 0; d < 3; ++d) {
    float size = fmaxf(reps[branch*3 + d], 1e-6f);
    float c = floorf(((float)idx[i*4+1+d] - startv[d]) / size);
    long long ci = (long long)c;
    if (ci < 0) ci = 0;
    if (ci > 4095) ci = 4095;
    cvox[i*3+d] = (int)ci;
    atomicMax(&mx[d], (int)ci);
  }
}

__global__ void k_ids(const int* __restrict__ idx, const int* __restrict__ cvox,
                      const int* __restrict__ mx, long long* __restrict__ ids)
{
  int i = blockIdx.x * blockDim.x + threadIdx.x;
  if (i >= NPTS) return;
  long long m0 = mx[0]+1, m1 = mx[1]+1, m2 = mx[2]+1;
  long long b = idx[i*4];
  ids[i] = ((b*m0 + cvox[i*3])*m1 + cvox[i*3+1])*m2 + cvox[i*3+2];
}

__global__ void k_uniq_rep(const long long* __restrict__ ids, int* __restrict__ repf)
{
  int i = blockIdx.x * blockDim.x + threadIdx.x;
  if (i >= NPTS) return;
  long long v = ids[i];
  int rep = 1;
  for (int j = 0; j < i; ++j) if (ids[j] == v) { rep = 0; break; }
  repf[i] = rep;
}

__global__ void k_uniq_inv(const long long* __restrict__ ids, const int* __restrict__ repf,
                           int* __restrict__ seg)
{
  int i = blockIdx.x * blockDim.x + threadIdx.x;
  if (i >= NPTS) return;
  long long v = ids[i];
  int c = 0;
  for (int j = 0; j < NPTS; ++j) c += (repf[j] && ids[j] < v);
  seg[i] = c;
}

__global__ void k_seg_count(const int* __restrict__ seg, float* __restrict__ cnt)
{
  int i = blockIdx.x * blockDim.x + threadIdx.x;
  if (i >= NPTS) return;
  atomicAdd(&cnt[seg[i]], 1.f);
}

__global__ void k_seg_sum(const float* __restrict__ X, const int* __restrict__ seg,
                          float* __restrict__ S)
{
  int t = blockIdx.x * blockDim.x + threadIdx.x;
  if (t >= NPTS*CCH) return;
  int i = t / CCH, c = t - i*CCH;
  atomicAdd(&S[(size_t)seg[i]*CCH + c], X[t]);
}

__global__ void k_sub_segmean(float* __restrict__ pw, const float* __restrict__ S,
                              const float* __restrict__ cnt, const int* __restrict__ seg)
{
  int t = blockIdx.x * blockDim.x + threadIdx.x;
  if (t >= NPTS*CCH) return;
  int i = t / CCH, c = t - i*CCH;
  int s = seg[i];
  pw[t] -= S[(size_t)s*CCH + c] / fmaxf(cnt[s], 1.f);
}

__global__ void k_max_part(const float* __restrict__ X, int n, float* __restrict__ part)
{
  __shared__ float sm[256];
  float m = -3.0e38f;
  for (int t = blockIdx.x*blockDim.x + threadIdx.x; t < n; t += gridDim.x*blockDim.x)
    m = fmaxf(m, X[t]);
  sm[threadIdx.x] = m; __syncthreads();
  for (int s = 128; s > 0; s >>= 1) {
    if (threadIdx.x < s) sm[threadIdx.x] = fmaxf(sm[threadIdx.x], sm[threadIdx.x+s]);
    __syncthreads();
  }
  if (threadIdx.x == 0) part[blockIdx.x] = sm[0];
}

__global__ void k_max_fin(const float* __restrict__ part, int n, float* __restrict__ out)
{
  __shared__ float sm[64];
  float m = (threadIdx.x < n) ? part[threadIdx.x] : -3.0e38f;
  sm[threadIdx.x] = m; __syncthreads();
  for (int s = 32; s > 0; s >>= 1) {
    if (threadIdx.x < s) sm[threadIdx.x] = fmaxf(sm[threadIdx.x], sm[threadIdx.x+s]);
    __syncthreads();
  }
  if (threadIdx.x == 0) out[0] = sm[0];
}

__global__ void k_exp(float* __restrict__ pw, const float* __restrict__ mv, int n)
{
  int t = blockIdx.x*blockDim.x + threadIdx.x;
  if (t >= n) return;
  pw[t] = expf(pw[t] - mv[0]);
}

__global__ void k_div_seg(float* __restrict__ pw, const float* __restrict__ S,
                          const int* __restrict__ seg)
{
  int t = blockIdx.x*blockDim.x + threadIdx.x;
  if (t >= NPTS*CCH) return;
  int i = t / CCH, c = t - i*CCH;
  pw[t] /= (S[(size_t)seg[i]*CCH + c] + 1e-6f);
}

__global__ void k_mul(float* __restrict__ a, const float* __restrict__ b, int n)
{
  int t = blockIdx.x*blockDim.x + threadIdx.x;
  if (t >= n) return;
  a[t] *= b[t];
}

__global__ void k_gather_seg(const float* __restrict__ S, const int* __restrict__ seg,
                             float* __restrict__ out)
{
  int t = blockIdx.x*blockDim.x + threadIdx.x;
  if (t >= NPTS*CCH) return;
  int i = t / CCH, c = t - i*CCH;
  out[t] = S[(size_t)seg[i]*CCH + c];
}

__global__ void k_adp_agg(const float* __restrict__ feat, const float* __restrict__ aw,
                          const float* __restrict__ br, float* __restrict__ agg)
{
  int i = blockIdx.x*blockDim.x + threadIdx.x;
  if (i >= NPTS) return;
  float l[3];
  for (int r = 0; r < 3; ++r) {
    float s = 0.f;
    for (int c = 0; c < CCH; ++c) s += feat[(size_t)i*CCH + c] * aw[r*CCH + c];
    l[r] = s;
  }
  float m = fmaxf(l[0], fmaxf(l[1], l[2]));
  float e0 = expf(l[0]-m), e1 = expf(l[1]-m), e2 = expf(l[2]-m);
  float inv = 1.f/(e0+e1+e2);
  e0*=inv; e1*=inv; e2*=inv;
  const size_t NC = (size_t)NPTS*CCH;
  for (int c = 0; c < CCH; ++c) {
    size_t t = (size_t)i*CCH + c;
    agg[t] = e0*br[t] + e1*br[NC+t] + e2*br[2*NC+t];
  }
}

__global__ void k_concat(const float* __restrict__ a, const float* __restrict__ b,
                         float* __restrict__ out)
{
  int t = blockIdx.x*blockDim.x + threadIdx.x;
  if (t >= NPTS*2*CCH) return;
  int i = t / (2*CCH), c = t - i*2*CCH;
  out[t] = (c < CCH) ? a[(size_t)i*CCH + c] : b[(size_t)i*CCH + (c - CCH)];
}

__global__ void k_scatter(const float* __restrict__ f, const int* __restrict__ idx,
                          float* __restrict__ grid)
{
  int t = blockIdx.x*blockDim.x + threadIdx.x;
  if (t >= NPTS*CCH) return;
  int i = t / CCH, c = t - i*CCH;
  int b = idx[i*4], z = idx[i*4+1], y = idx[i*4+2], xw = idx[i*4+3];
  grid[((((size_t)b*DD + z)*HH + y)*WW + xw)*CCH + c] = f[t];
}

// ============================================================================
// Host side
// ============================================================================
static inline int eb(size_t n) { return (int)((n + 255) / 256); }

static void gemm(const float* X, int ldx, const float* Wt, int ldw, const float* b,
                 float* Y, int ldy, int nr, int mc, int kd, int act, hipStream_t s)
{
  int waves = nr / 16;
  int blocks = (waves + 3) / 4;
  if (kd == 96 && mc == 96)
    k_gemm_strip<96,6><<<blocks, 128, 0, s>>>(X, ldx, Wt, ldw, b, Y, ldy, nr, act);
  else if (kd == 96 && mc == 48)
    k_gemm_strip<96,3><<<blocks, 128, 0, s>>>(X, ldx, Wt, ldw, b, Y, ldy, nr, act);
  else if (kd == 96 && mc == 64)
    k_gemm_strip<96,4><<<blocks, 128, 0, s>>>(X, ldx, Wt, ldw, b, Y, ldy, nr, act);
  else  // kd == 192, mc == 96
    k_gemm_strip<192,6><<<blocks, 128, 0, s>>>(X, ldx, Wt, ldw, b, Y, ldy, nr, act);
}

static void bn(const float* Z, const float* g, const float* be, const float* rpre,
               const float* rpost, float* out, int nr, int mc, int act,
               float* mean, float* var, hipStream_t s)
{
  k_colstats<<<mc, 256, 0, s>>>(Z, mc, nr, mc, mean, var);
  size_t tot = (size_t)nr * mc;
  k_bn_act<<<eb(tot), 256, 0, s>>>(Z, mean, var, g, be, rpre, rpost, out, (int)tot, mc, act);
}

static void conv(const float* grid, const int* idx, const float* w,
                 float* part, float* out, hipStream_t s)
{
  const size_t NC = (size_t)NPTS * CCH;
  int waves = NPTS / 16;
  int blocks = (waves + 3) / 4;
  k_conv_strip<<<blocks, 128, 0, s>>>(grid, idx, w, part,          0,  9);
  k_conv_strip<<<blocks, 128, 0, s>>>(grid, idx, w, part + NC,     9, 18);
  k_conv_strip<<<blocks, 128, 0, s>>>(grid, idx, w, part + 2*NC,  18, 27);
  k_sum3<<<eb(NC), 256, 0, s>>>(part, part + NC, part + 2*NC, out, (int)NC);
}

extern "C" void kernel_launch(void* const* d_in, const int* in_sizes, int n_in,
                              void* d_out, int out_size, void* d_ws, size_t ws_size,
                              hipStream_t stream)
{
  (void)in_sizes; (void)n_in; (void)out_size; (void)ws_size;
  const float* x      = (const float*)d_in[0];
  const int*   indices= (const int*)  d_in[1];
  const float* fp_w   = (const float*)d_in[2];
  const float* fp_b   = (const float*)d_in[3];
  const float* fp_g   = (const float*)d_in[4];
  const float* fp_be  = (const float*)d_in[5];
  const float* att_w1 = (const float*)d_in[6];
  const float* att_b1 = (const float*)d_in[7];
  const float* att_w2 = (const float*)d_in[8];
  const float* att_b2 = (const float*)d_in[9];
  const float* ff_w1  = (const float*)d_in[10];
  const float* ff_b1  = (const float*)d_in[11];
  const float* ff_g   = (const float*)d_in[12];
  const float* ff_be  = (const float*)d_in[13];
  const float* ff_w2  = (const float*)d_in[14];
  const float* ff_b2  = (const float*)d_in[15];
  const float* sa_w1  = (const float*)d_in[16];
  const float* sa_b1  = (const float*)d_in[17];
  const float* sa_w2  = (const float*)d_in[18];
  const float* sa_b2  = (const float*)d_in[19];
  const float* fj_w1  = (const float*)d_in[20];
  const float* fj_b1  = (const float*)d_in[21];
  const float* fj_g   = (const float*)d_in[22];
  const float* fj_be  = (const float*)d_in[23];
  const float* fj_w2  = (const float*)d_in[24];
  const float* fj_b2  = (const float*)d_in[25];
  const float* proj_w = (const float*)d_in[26];
  const float* proj_g = (const float*)d_in[27];
  const float* proj_be= (const float*)d_in[28];
  const float* lw_w   = (const float*)d_in[29];
  const float* lw_g   = (const float*)d_in[30];
  const float* lw_be  = (const float*)d_in[31];
  const float* w_w    = (const float*)d_in[32];
  const float* adp_w  = (const float*)d_in[33];
  const float* fuse_w = (const float*)d_in[34];
  const float* fuse_g = (const float*)d_in[35];
  const float* fuse_be= (const float*)d_in[36];
  const float* conv1_w= (const float*)d_in[37];
  const float* bn1_g  = (const float*)d_in[38];
  const float* bn1_be = (const float*)d_in[39];
  const float* conv2_w= (const float*)d_in[40];
  const float* bn2_g  = (const float*)d_in[41];
  const float* bn2_be = (const float*)d_in[42];

  float* wsf = (float*)d_ws;
  size_t off = 0;
  auto F = [&](size_t n) -> float* { float* r = wsf + off; off += n; return r; };
  const size_t NC = (size_t)NPTS * CCH;

  float* z9    = F((size_t)NPTS*9);
  float* p     = F((size_t)NPTS*9);
  float* enh   = F((size_t)NPTS*9);
  float* bufA  = F(NC);
  float* bufB  = F(NC);
  float* fu    = F(NC);
  float* feat  = F(NC);
  float* pw    = F(NC);
  float* pf    = F(NC);
  float* S     = F(NC);
  float* br    = F(3*NC);     // branch feats; reused as conv partials after agg
  float* agg   = F(NC);
  float* last  = F(NC);
  float* cat   = F(2*NC);
  float* fused = F(NC);
  float* f1    = F(NC);
  float* yb    = F(NC);
  float* mean  = F(256);
  float* var   = F(256);
  float* startv= F(4);
  float* reps  = F(12);
  float* mpart = F(256);
  float* maxv  = F(4);
  float* gm    = F(NPTS);
  float* gs    = F(3*NPTS);
  float* lind  = F(2*NPTS);
  float* cnt   = F(NPTS);
  int*   seg   = (int*)F(NPTS);
  int*   rnk   = (int*)F(NPTS);
  int*   repf  = (int*)F(NPTS);
  int*   cvox  = (int*)F(3*NPTS);
  int*   mx    = (int*)F(4);
  off = (off + 1) & ~(size_t)1;               // 8-byte align for int64 ids
  long long* ids = (long long*)F(2*NPTS);
  float* grid  = F((size_t)BB*DD*HH*WW*CCH);
  const size_t gridBytes = (size_t)BB*DD*HH*WW*CCH*sizeof(float);

  // ---- front: p = relu(bn(x @ fp_w^T + fp_b)) ----
  k_small_gemm<<<eb((size_t)NPTS*9), 256, 0, stream>>>(x, CCH, fp_w, CCH, fp_b, z9, 9, NPTS, 9, CCH);
  bn(z9, fp_g, fp_be, nullptr, nullptr, p, NPTS, 9, 1, mean, var, stream);
  k_att_enh<<<eb(NPTS), 256, 0, stream>>>(p, att_w1, att_b1, att_w2, att_b2, enh);

  // fu = relu(bn(enh @ ff_w1^T + ff_b1)) @ ff_w2^T + ff_b2
  k_small_gemm<<<eb(NC), 256, 0, stream>>>(enh, 9, ff_w1, 9, ff_b1, bufA, CCH, NPTS, CCH, 9);
  bn(bufA, ff_g, ff_be, nullptr, nullptr, bufB, NPTS, CCH, 1, mean, var, stream);
  gemm(bufB, CCH, ff_w2, CCH, ff_b2, fu, CCH, NPTS, CCH, CCH, 0, stream);

  // sem / feat blend
  gemm(fu, CCH, sa_w1, CCH, sa_b1, bufA, 48, NPTS, 48, CCH, 1, stream);
  k_sem_feat<<<eb(NPTS), 256, 0, stream>>>(bufA, sa_w2, sa_b2, fu, x, feat);

  // geometry
  k_knn<<<eb(NPTS), 256, 0, stream>>>(indices, lind);
  k_coord_min<<<1, 256, 0, stream>>>(indices, startv);

  // fl = relu(bn(feat @ fj_w1^T + fj_b1)) @ fj_w2^T + fj_b2 ; then grid sizes
  gemm(feat, CCH, fj_w1, CCH, fj_b1, bufA, 64, NPTS, 64, CCH, 0, stream);
  bn(bufA, fj_g, fj_be, nullptr, nullptr, bufB, NPTS, 64, 1, mean, var, stream);
  k_small_gemm<<<eb((size_t)NPTS*3), 256, 0, stream>>>(bufB, 64, fj_w2, 64, fj_b2, pf, 3, NPTS, 3, 64);
  k_gridsizes<<<eb(NPTS), 256, 0, stream>>>(pf, lind, gs, gm);
  k_rank<<<eb(NPTS), 256, 0, stream>>>(gm, rnk);
  hipMemsetAsync(reps, 0, 12*sizeof(float), stream);
  k_reps<<<eb(NPTS), 256, 0, stream>>>(gs, rnk, reps);

  // ---- 3 cluster branches ----
  for (int i = 0; i < 3; ++i) {
    hipMemsetAsync(mx, 0, 3*sizeof(int), stream);
    k_voxel<<<eb(NPTS), 256, 0, stream>>>(indices, startv, reps, i, cvox, mx);
    k_ids<<<eb(NPTS), 256, 0, stream>>>(indices, cvox, mx, ids);
    k_uniq_rep<<<eb(NPTS), 256, 0, stream>>>(ids, repf);
    k_uniq_inv<<<eb(NPTS), 256, 0, stream>>>(ids, repf, seg);
    hipMemsetAsync(cnt, 0, NPTS*sizeof(float), stream);
    k_seg_count<<<eb(NPTS), 256, 0, stream>>>(seg, cnt);

    gemm(feat, CCH, lw_w + (size_t)i*CCH*CCH, CCH, nullptr, bufA, CCH, NPTS, CCH, CCH, 0, stream);
    bn(bufA, lw_g + i*CCH, lw_be + i*CCH, nullptr, nullptr, pw, NPTS, CCH, 1, mean, var, stream);
    hipMemsetAsync(S, 0, NC*sizeof(float), stream);
    k_seg_sum<<<eb(NC), 256, 0, stream>>>(pw, seg, S);
    k_sub_segmean<<<eb(NC), 256, 0, stream>>>(pw, S, cnt, seg);

    gemm(pw, CCH, w_w + (size_t)i*CCH*CCH, CCH, nullptr, bufA, CCH, NPTS, CCH, CCH, 0, stream);
    k_max_part<<<64, 256, 0, stream>>>(bufA, (int)NC, mpart);
    k_max_fin<<<1, 64, 0, stream>>>(mpart, 64, maxv);
    k_exp<<<eb(NC), 256, 0, stream>>>(bufA, maxv, (int)NC);
    hipMemsetAsync(S, 0, NC*sizeof(float), stream);
    k_seg_sum<<<eb(NC), 256, 0, stream>>>(bufA, seg, S);
    k_div_seg<<<eb(NC), 256, 0, stream>>>(bufA, S, seg);

    gemm(feat, CCH, proj_w + (size_t)i*CCH*CCH, CCH, nullptr, bufB, CCH, NPTS, CCH, CCH, 0, stream);
    bn(bufB, proj_g + i*CCH, proj_be + i*CCH, nullptr, nullptr, pf, NPTS, CCH, 1, mean, var, stream);
    k_mul<<<eb(NC), 256, 0, stream>>>(pf, bufA, (int)NC);
    hipMemsetAsync(S, 0, NC*sizeof(float), stream);
    k_seg_sum<<<eb(NC), 256, 0, stream>>>(pf, seg, S);
    k_gather_seg<<<eb(NC), 256, 0, stream>>>(S, seg, br + (size_t)i*NC);
  }

  // adaptive aggregation + fuse
  k_adp_agg<<<eb(NPTS), 256, 0, stream>>>(feat, adp_w, br, agg);
  gemm(feat, CCH, proj_w + (size_t)3*CCH*CCH, CCH, nullptr, bufA, CCH, NPTS, CCH, CCH, 0, stream);
  bn(bufA, proj_g + 3*CCH, proj_be + 3*CCH, nullptr, nullptr, last, NPTS, CCH, 1, mean, var, stream);
  k_concat<<<eb(2*NC), 256, 0, stream>>>(last, agg, cat);
  gemm(cat, 2*CCH, fuse_w, 2*CCH, nullptr, bufA, CCH, NPTS, CCH, 2*CCH, 0, stream);
  bn(bufA, fuse_g, fuse_be, nullptr, feat, fused, NPTS, CCH, 1, mean, var, stream); // relu(bn) + feat

  // ---- conv1 ----  (br is free now; reuse as 3 tap-group partial buffers)
  hipMemsetAsync(grid, 0, gridBytes, stream);
  k_scatter<<<eb(NC), 256, 0, stream>>>(fused, indices, grid);
  conv(grid, indices, conv1_w, br, yb, stream);
  bn(yb, bn1_g, bn1_be, nullptr, nullptr, f1, NPTS, CCH, 1, mean, var, stream);

  // ---- conv2 + residual ----
  hipMemsetAsync(grid, 0, gridBytes, stream);
  k_scatter<<<eb(NC), 256, 0, stream>>>(f1, indices, grid);
  conv(grid, indices, conv2_w, br, yb, stream);
  // out = relu(bn(yb) + fused)
  bn(yb, bn2_g, bn2_be, fused, nullptr, (float*)d_out, NPTS, CCH, 1, mean, var, stream);
}